// PyrDown_4690104287654
// MI455X (gfx1250) — compile-verified
//
#include <hip/hip_runtime.h>

#define AS1 __attribute__((address_space(1)))
#define AS3 __attribute__((address_space(3)))

#if defined(__HIP_DEVICE_COMPILE__) && __has_builtin(__builtin_amdgcn_global_load_async_to_lds_b32)
#define USE_ASYNC_LDS 1
#else
#define USE_ASYNC_LDS 0
#endif

// ---------------- geometry (fixed by the reference problem) ----------------
#define N_IMG   48          // B*C = 16*3
#define IMG_H   1024
#define IMG_W   1024
#define OUT_H   512
#define OUT_W   512
#define OTW     32          // output tile width
#define OTH     32          // output tile height
#define BW      (2*OTW + 2) // 66 blur cols needed by the 4-tap downsample
#define BH      (2*OTH + 2) // 66 blur rows
#define TIW     (BW + 4)    // 70 raw input cols (5-tap halo)
#define TIH     (BH + 4)    // 70 raw input rows
#define SRAW    72          // LDS row stride for raw tiles
#define SCNV    68          // LDS row stride for conv tiles
#define NTHD    256

// LDS layout (floats) inside one dynamic allocation
#define OFF_IN  0
#define OFF_MK  (OFF_IN + TIH * SRAW)          // 5040
#define OFF_HN  (OFF_MK + TIH * SRAW)          // +5040
#define OFF_HD  (OFF_HN + TIH * SCNV)          // +4760
#define OFF_BL  (OFF_HD + TIH * SCNV)          // +4760
#define OFF_DS  (OFF_BL + BH * SCNV)           // +4488
#define LDS_FLOATS (OFF_DS + BH * OTW)         // +2112 = 26200 floats
#define LDS_BYTES  (LDS_FLOATS * 4)            // 104800 B (< 320 KB/WGP; 3 WGs/WGP)

__device__ __forceinline__ int refl(int i, int n) {
  // jnp.pad 'reflect' with pad<=2: -1->1, -2->2, n->n-2, n+1->n-3
  i = (i < 0) ? -i : i;
  return (i >= n) ? (2 * n - 2 - i) : i;
}

__device__ __forceinline__ void cp_async_f32(const float* g, float* l) {
#if USE_ASYNC_LDS
  // global_load_async_to_lds_b32: per-lane global addr -> per-lane LDS addr,
  // tracked on ASYNCcnt (no VGPR round trip).
  __builtin_amdgcn_global_load_async_to_lds_b32(
      (AS1 int*)const_cast<float*>(g), (AS3 int*)l, /*offset=*/0, /*cpol=*/0);
#else
  *l = *g;
#endif
}

__device__ __forceinline__ void wait_async0() {
#if USE_ASYNC_LDS
#if __has_builtin(__builtin_amdgcn_s_wait_asynccnt)
  __builtin_amdgcn_s_wait_asynccnt(0);
#else
  asm volatile("s_wait_asynccnt 0" ::: "memory");
#endif
#endif
}

__global__ __launch_bounds__(NTHD) void pyrdown_masked_blur_half(
    const float* __restrict__ xin, const float* __restrict__ msk,
    float* __restrict__ out) {
  extern __shared__ float smem[];
  float* s_in = smem + OFF_IN;   // raw input tile
  float* s_mk = smem + OFF_MK;   // raw mask tile
  float* s_hn = smem + OFF_HN;   // horizontal conv of x*m
  float* s_hd = smem + OFF_HD;   // horizontal conv of m
  float* s_bl = smem + OFF_BL;   // normalized blur
  float* s_ds = smem + OFF_DS;   // horizontally downsampled blur

  const int tid = threadIdx.x;
  const int img = blockIdx.z;
  const int OX0 = blockIdx.x * OTW;
  const int OY0 = blockIdx.y * OTH;
  // raw-tile origin in image coords (blur col 2*OX0-1 needs input col 2*OX0-3)
  const int IR0 = 2 * OY0 - 3;
  const int IC0 = 2 * OX0 - 3;

  const float* gx = xin + (size_t)img * (IMG_H * IMG_W);
  const float* gm = msk + (size_t)img * (IMG_H * IMG_W);

  // -------- Stage A: async-stage raw tiles (reflect halo) into LDS --------
  for (int t = tid; t < TIH * TIW; t += NTHD) {
    int r = t / TIW, c = t - r * TIW;
    size_t off = (size_t)refl(IR0 + r, IMG_H) * IMG_W + refl(IC0 + c, IMG_W);
    cp_async_f32(gx + off, &s_in[r * SRAW + c]);
    cp_async_f32(gm + off, &s_mk[r * SRAW + c]);
  }
  wait_async0();
  __syncthreads();

  // -------- Stage B: horizontal 5-tap [1 4 6 4 1]/16 on (x*m) and m --------
  const float w0 = 1.f / 16.f, w1 = 4.f / 16.f, w2 = 6.f / 16.f;
  // 4 outputs per task from 8 loads per array (amortize LDS reads)
  for (int t = tid; t < TIH * 17; t += NTHD) {
    int r = t / 17;
    int c0 = (t - r * 17) * 4;
    float mv[8], pv[8];
#pragma unroll
    for (int q = 0; q < 8; ++q) {
      float xv = s_in[r * SRAW + c0 + q];
      mv[q] = s_mk[r * SRAW + c0 + q];
      pv[q] = xv * mv[q];
    }
#pragma unroll
    for (int o = 0; o < 4; ++o) {
      int c = c0 + o;
      if (c < BW) {
        s_hn[r * SCNV + c] = w0 * (pv[o] + pv[o + 4]) + w1 * (pv[o + 1] + pv[o + 3]) + w2 * pv[o + 2];
        s_hd[r * SCNV + c] = w0 * (mv[o] + mv[o + 4]) + w1 * (mv[o + 1] + mv[o + 3]) + w2 * mv[o + 2];
      }
    }
  }
  __syncthreads();

  // -------- Stage C: vertical 5-tap + masked normalization --------
  for (int t = tid; t < BH * BW; t += NTHD) {
    int i = t / BW, j = t - i * BW;
    const float* hn = &s_hn[i * SCNV + j];
    const float* hd = &s_hd[i * SCNV + j];
    float n = w0 * (hn[0] + hn[4 * SCNV]) + w1 * (hn[SCNV] + hn[3 * SCNV]) + w2 * hn[2 * SCNV];
    float d = w0 * (hd[0] + hd[4 * SCNV]) + w1 * (hd[SCNV] + hd[3 * SCNV]) + w2 * hd[2 * SCNV];
    d = (d == 0.0f) ? 1.0f : d;   // den + (den == 0)
    s_bl[i * SCNV + j] = n / d;
  }
  __syncthreads();

  // -------- Stage D1: horizontal [1 3 3 1]/8 downsample --------
  for (int t = tid; t < BH * OTW; t += NTHD) {
    int i = t >> 5, u = t & 31;
    const float* b = &s_bl[i * SCNV + 2 * u];
    s_ds[i * OTW + u] = 0.125f * (b[0] + 3.f * b[1] + 3.f * b[2] + b[3]);
  }
  __syncthreads();

  // -------- Stage D2: vertical [1 3 3 1]/8 downsample + coalesced store ----
  float* go = out + (size_t)img * (OUT_H * OUT_W);
  for (int t = tid; t < OTH * OTW; t += NTHD) {
    int v = t >> 5, u = t & 31;
    const float* d = &s_ds[(2 * v) * OTW + u];
    float r = 0.125f * (d[0] + 3.f * d[OTW] + 3.f * d[2 * OTW] + d[3 * OTW]);
    go[(size_t)(OY0 + v) * OUT_W + (OX0 + u)] = r;
  }
}

extern "C" void kernel_launch(void* const* d_in, const int* in_sizes, int n_in,
                              void* d_out, int out_size, void* d_ws, size_t ws_size,
                              hipStream_t stream) {
  (void)in_sizes; (void)n_in; (void)d_ws; (void)ws_size; (void)out_size;
  const float* x = (const float*)d_in[0];
  const float* m = (const float*)d_in[1];
  float* o = (float*)d_out;
  dim3 grid(OUT_W / OTW, OUT_H / OTH, N_IMG);  // 16 x 16 x 48 = 12288 tiles
  pyrdown_masked_blur_half<<<grid, dim3(NTHD, 1, 1), LDS_BYTES, stream>>>(x, m, o);
}